// RecurrentGCN_50362786513548
// MI455X (gfx1250) — compile-verified
//
#include <hip/hip_runtime.h>
#include <math.h>

typedef float v2f __attribute__((ext_vector_type(2)));
typedef float v8f __attribute__((ext_vector_type(8)));

#define N_NODES 50000
#define N_EDGES 800000
#define T_STEPS 12
#define HIDDEN  128
#define FDIM    36      // 35 stacked features padded to 36 (9 x K=4 WMMA steps)
#define CIN     135

// ---------------------------------------------------------------- utilities
__global__ void zero_f32(float* __restrict__ p, int n) {
    int i = blockIdx.x * blockDim.x + threadIdx.x;
    if (i < n) p[i] = 0.0f;
}

// ---------------------------------------------------------------- degrees
__global__ void deg_kernel(const int* __restrict__ ei, const float* __restrict__ ew,
                           float* __restrict__ degO, float* __restrict__ degI) {
    int e = blockIdx.x * blockDim.x + threadIdx.x;
    if (e < N_EDGES) {
        int r = ei[e];
        int c = ei[N_EDGES + e];
        float w = ew[e];
        atomicAdd(&degO[r], w);
        atomicAdd(&degI[c], w);
    }
}

// invert degrees in place, seed x_prev and predictions[:,0]
__global__ void init_nodes(const float* __restrict__ x,
                           float* __restrict__ invO, float* __restrict__ invI,
                           float* __restrict__ xprev, float* __restrict__ out) {
    int n = blockIdx.x * blockDim.x + threadIdx.x;
    if (n < N_NODES) {
        float d0 = invO[n];
        float d1 = invI[n];
        invO[n] = (d0 > 0.0f) ? 1.0f / d0 : 0.0f;
        invI[n] = (d1 > 0.0f) ? 1.0f / d1 : 0.0f;
        float x0 = x[n * T_STEPS];           // x[:,0]
        xprev[n] = x0;
        out[n * (T_STEPS + 1)] = x0;         // predictions[:,0]
    }
}

// ---------------------------------------------------------------- weight stacking
// W shape (2, 3, 135, 128). Only first 7 input rows matter (hidden input is 0).
// Stacked 36x128: [W00+W10 ; W01 ; W11 ; W02 ; W12 ; zero-pad row]
__device__ __forceinline__ int widx(int d, int k, int c, int h) {
    return ((d * 3 + k) * CIN + c) * HIDDEN + h;
}

__global__ void stack_weights(const float* __restrict__ Wz, const float* __restrict__ Wh,
                              float* __restrict__ Wzs, float* __restrict__ Whs) {
    int i = blockIdx.x * blockDim.x + threadIdx.x;
    if (i >= FDIM * HIDDEN) return;
    int h = i & (HIDDEN - 1);
    int r = i >> 7;
    float vz = 0.0f, vh = 0.0f;
    if (r < 7) {
        vz = Wz[widx(0, 0, r, h)] + Wz[widx(1, 0, r, h)];
        vh = Wh[widx(0, 0, r, h)] + Wh[widx(1, 0, r, h)];
    } else if (r < 14) {
        vz = Wz[widx(0, 1, r - 7, h)];  vh = Wh[widx(0, 1, r - 7, h)];
    } else if (r < 21) {
        vz = Wz[widx(1, 1, r - 14, h)]; vh = Wh[widx(1, 1, r - 14, h)];
    } else if (r < 28) {
        vz = Wz[widx(0, 2, r - 21, h)]; vh = Wh[widx(0, 2, r - 21, h)];
    } else if (r < 35) {
        vz = Wz[widx(1, 2, r - 28, h)]; vh = Wh[widx(1, 2, r - 28, h)];
    }
    Wzs[i] = vz;
    Whs[i] = vh;
}

// ---------------------------------------------------------------- per-step feature build
// F[n, 0:7] = [x_prev, env[n,0:4,t], coords[n,0:2]]
__global__ void build_x(const float* __restrict__ xprev, const float* __restrict__ env,
                        const float* __restrict__ coords, float* __restrict__ F, int t) {
    int n = blockIdx.x * blockDim.x + threadIdx.x;
    if (n < N_NODES) {
        float* f = F + n * FDIM;
        f[0] = xprev[n];
        f[1] = env[n * 48 + 0 * T_STEPS + t];
        f[2] = env[n * 48 + 1 * T_STEPS + t];
        f[3] = env[n * 48 + 2 * T_STEPS + t];
        f[4] = env[n * 48 + 3 * T_STEPS + t];
        f[5] = coords[2 * n];
        f[6] = coords[2 * n + 1];
    }
}

// edge scatter: accO[row] += w * F[col, offO:offO+7]; accI[col] += w * F[row, offI:offI+7]
__global__ void edge_prop(const int* __restrict__ ei, const float* __restrict__ ew,
                          const float* __restrict__ F, int offO, int offI,
                          float* __restrict__ accO, float* __restrict__ accI) {
    int e = blockIdx.x * blockDim.x + threadIdx.x;
    if (e < N_EDGES) {
        int r = ei[e];
        int c = ei[N_EDGES + e];
        float w = ew[e];
        const float* so = F + c * FDIM + offO;
        const float* si = F + r * FDIM + offI;
        float* ao = accO + r * 7;
        float* ai = accI + c * 7;
#pragma unroll
        for (int f = 0; f < 7; ++f) {
            atomicAdd(&ao[f], w * so[f]);
            atomicAdd(&ai[f], w * si[f]);
        }
    }
}

// F[n,7:14] = invO*accO ; F[n,14:21] = invI*accI
__global__ void scale_first(const float* __restrict__ accO, const float* __restrict__ accI,
                            const float* __restrict__ invO, const float* __restrict__ invI,
                            float* __restrict__ F) {
    int n = blockIdx.x * blockDim.x + threadIdx.x;
    if (n < N_NODES) {
        float io = invO[n], ii = invI[n];
        float* f = F + n * FDIM;
#pragma unroll
        for (int k = 0; k < 7; ++k) {
            f[7 + k]  = io * accO[n * 7 + k];
            f[14 + k] = ii * accI[n * 7 + k];
        }
    }
}

// F[n,21:28] = 2*invO*accO - X ; F[n,28:35] = 2*invI*accI - X ; F[n,35]=0
__global__ void combine_second(const float* __restrict__ accO, const float* __restrict__ accI,
                               const float* __restrict__ invO, const float* __restrict__ invI,
                               float* __restrict__ F) {
    int n = blockIdx.x * blockDim.x + threadIdx.x;
    if (n < N_NODES) {
        float io = invO[n], ii = invI[n];
        float* f = F + n * FDIM;
#pragma unroll
        for (int k = 0; k < 7; ++k) {
            float x0 = f[k];
            f[21 + k] = 2.0f * io * accO[n * 7 + k] - x0;
            f[28 + k] = 2.0f * ii * accI[n * 7 + k] - x0;
        }
        f[35] = 0.0f;
    }
}

// ---------------------------------------------------------------- fused WMMA GEMM + GRU + projection
// Block = 256 threads = 8 waves. Block handles 16 rows x 128 cols.
// Wave w handles cols [16w, 16w+16). Two accumulators: Z-gate and H-gate.
// K = 36 = 9 WMMA_F32_16X16X4_F32 steps per accumulator.
__global__ __launch_bounds__(256) void gemm_gru(
    const float* __restrict__ F,
    const float* __restrict__ Wzs, const float* __restrict__ Whs,
    const float* __restrict__ bz,  const float* __restrict__ bh,
    const float* __restrict__ linW, const float* __restrict__ linb,
    const float* __restrict__ night,
    float* __restrict__ out, float* __restrict__ xprev, int t)
{
    __shared__ float lds[16 * HIDDEN];

    int tid  = threadIdx.x;
    int wave = tid >> 5;
    int lane = tid & 31;
    int m    = lane & 15;
    int kh   = lane >> 4;        // half-wave select
    int col0 = wave * 16;
    int row0 = blockIdx.x * 16;  // N = 3125 * 16 exactly
    int gm   = row0 + m;
    int col  = col0 + m;

    v8f cz = {0.f, 0.f, 0.f, 0.f, 0.f, 0.f, 0.f, 0.f};
    v8f ch = {0.f, 0.f, 0.f, 0.f, 0.f, 0.f, 0.f, 0.f};

    // A frag: lane m row, VGPR0 = K (2*kh), VGPR1 = K (2*kh+1)
    const float* Arow = F   + gm * FDIM + 2 * kh;
    const float* Bz   = Wzs + (2 * kh) * HIDDEN + col;
    const float* Bh   = Whs + (2 * kh) * HIDDEN + col;

#pragma unroll
    for (int ks = 0; ks < 9; ++ks) {
        int k0 = ks * 4;
        v2f a;   a.x   = Arow[k0];            a.y   = Arow[k0 + 1];
        v2f vbz; vbz.x = Bz[k0 * HIDDEN];     vbz.y = Bz[(k0 + 1) * HIDDEN];
        v2f vbh; vbh.x = Bh[k0 * HIDDEN];     vbh.y = Bh[(k0 + 1) * HIDDEN];
        cz = __builtin_amdgcn_wmma_f32_16x16x4_f32(false, a, false, vbz, (short)0, cz, false, false);
        ch = __builtin_amdgcn_wmma_f32_16x16x4_f32(false, a, false, vbh, (short)0, ch, false, false);
    }

    float bzc = bz[col];
    float bhc = bh[col];
    float lwc = linW[col];

    // C layout: VGPR r -> rows (r, r+8) across half-waves, col = col0 + m
#pragma unroll
    for (int r = 0; r < 8; ++r) {
        float z  = 1.0f / (1.0f + __expf(-(cz[r] + bzc)));
        float ht = tanhf(ch[r] + bhc);
        float hn = (1.0f - z) * ht;               // hid0 == 0 => H_new = (1-Z)*H~
        float p  = (hn > 0.0f ? hn : 0.0f) * lwc; // relu then weight for projection
        lds[(r + kh * 8) * HIDDEN + col] = p;
    }
    __syncthreads();

    if (tid < 16) {
        float s = 0.0f;
#pragma unroll 8
        for (int cc = 0; cc < HIDDEN; ++cc) s += lds[tid * HIDDEN + cc];
        int n = row0 + tid;
        float o = (s + linb[0]) * night[n * (T_STEPS + 1) + t + 1];
        out[n * (T_STEPS + 1) + t + 1] = o;
        xprev[n] = o;
    }
}

// ---------------------------------------------------------------- host launch
extern "C" void kernel_launch(void* const* d_in, const int* in_sizes, int n_in,
                              void* d_out, int out_size, void* d_ws, size_t ws_size,
                              hipStream_t stream) {
    (void)in_sizes; (void)n_in; (void)out_size; (void)ws_size;

    const float* x      = (const float*)d_in[0];   // (N, T)
    const float* env    = (const float*)d_in[1];   // (N, 4, T)
    const float* coords = (const float*)d_in[2];   // (N, 2)
    const int*   ei     = (const int*)  d_in[3];   // (2, E)
    const float* ew     = (const float*)d_in[4];   // (E,)
    const float* night  = (const float*)d_in[5];   // (N, T+1)
    const float* Wz     = (const float*)d_in[6];   // (2,3,135,128)
    const float* bz     = (const float*)d_in[7];   // (128,)
    // d_in[8], d_in[9] (Wr, br) are unused: R gates a zero hidden state.
    const float* Wh     = (const float*)d_in[10];  // (2,3,135,128)
    const float* bh     = (const float*)d_in[11];  // (128,)
    const float* linW   = (const float*)d_in[12];  // (128,1)
    const float* linb   = (const float*)d_in[13];  // (1,)

    float* out = (float*)d_out;                    // (N, T+1)

    // workspace layout (floats)
    float* ws    = (float*)d_ws;
    float* F     = ws;                              // N*36
    float* accO  = F    + (size_t)N_NODES * FDIM;   // N*7
    float* accI  = accO + (size_t)N_NODES * 7;      // N*7
    float* invO  = accI + (size_t)N_NODES * 7;      // N
    float* invI  = invO + N_NODES;                  // N
    float* xprev = invI + N_NODES;                  // N
    float* Wzs   = xprev + N_NODES;                 // 36*128
    float* Whs   = Wzs + FDIM * HIDDEN;             // 36*128

    const int TPB        = 256;
    const int nodeBlocks = (N_NODES + TPB - 1) / TPB;       // 196
    const int edgeBlocks = (N_EDGES + TPB - 1) / TPB;       // 3125
    const int accBlocks  = (14 * N_NODES + TPB - 1) / TPB;  // 2735
    const int wBlocks    = (FDIM * HIDDEN + TPB - 1) / TPB; // 18
    const int gemmBlocks = N_NODES / 16;                    // 3125 (exact)

    // one-time (per launch) setup
    zero_f32<<<(2 * N_NODES + TPB - 1) / TPB, TPB, 0, stream>>>(invO, 2 * N_NODES);
    deg_kernel<<<edgeBlocks, TPB, 0, stream>>>(ei, ew, invO, invI);
    init_nodes<<<nodeBlocks, TPB, 0, stream>>>(x, invO, invI, xprev, out);
    stack_weights<<<wBlocks, TPB, 0, stream>>>(Wz, Wh, Wzs, Whs);

    for (int t = 0; t < T_STEPS; ++t) {
        build_x<<<nodeBlocks, TPB, 0, stream>>>(xprev, env, coords, F, t);

        // first-order propagation
        zero_f32<<<accBlocks, TPB, 0, stream>>>(accO, 14 * N_NODES);
        edge_prop<<<edgeBlocks, TPB, 0, stream>>>(ei, ew, F, 0, 0, accO, accI);
        scale_first<<<nodeBlocks, TPB, 0, stream>>>(accO, accI, invO, invI, F);

        // second-order (Chebyshev) propagation
        zero_f32<<<accBlocks, TPB, 0, stream>>>(accO, 14 * N_NODES);
        edge_prop<<<edgeBlocks, TPB, 0, stream>>>(ei, ew, F, 7, 14, accO, accI);
        combine_second<<<nodeBlocks, TPB, 0, stream>>>(accO, accI, invO, invI, F);

        // fused WMMA GEMM + GRU activations + linear projection + night gate
        gemm_gru<<<gemmBlocks, TPB, 0, stream>>>(F, Wzs, Whs, bz, bh, linW, linb,
                                                 night, out, xprev, t);
    }
}